// FSUSqrt_39127152067000
// MI455X (gfx1250) — compile-verified
//
#include <hip/hip_runtime.h>

// FSUSqrt (unipolar, emit=True, ENTRY_SR=2) bit-stream recurrence.
// T=64 sequential steps over N=524288 independent lanes.
// Memory-bound: ~274 MB total traffic -> ~12 us floor at 23.3 TB/s.
// CDNA5 path: 8-deep async global->LDS staging pipeline (ASYNCcnt) + NT stores.

typedef float v4f __attribute__((ext_vector_type(4)));

#define DEPTH 8
#define DEPTH_M1 7
#define BLOCK 256

#define STR2(x) #x
#define STR(x) STR2(x)

// Convert a generic pointer to a __shared__ object into a raw LDS byte offset
// (generic LDS addresses carry the offset in the low 32 bits; the optimizer
// folds the AS3->generic->AS3 round trip to a direct LDS address).
__device__ __forceinline__ unsigned lds_off(const void* p) {
  return (unsigned)(unsigned long long)
         (__attribute__((address_space(3))) const void*)p;
}

// GLOBAL_LOAD_ASYNC_TO_LDS_B128: dest = LDS byte address (VGPR),
// src = 64-bit global address (VGPR pair), GV mode (saddr = off).
// Tracked by ASYNCcnt, not LOADcnt.
__device__ __forceinline__ void async_copy_b128(unsigned ldsoff,
                                                const void* gaddr) {
  asm volatile("global_load_async_to_lds_b128 %0, %1, off"
               :
               : "v"(ldsoff), "v"((unsigned long long)gaddr)
               : "memory");
}

// One FSUSqrt step on a single {0,1}-bit lane.
__device__ __forceinline__ float fsu_step(float xb, float& acc, float& emit,
                                          float& sr0) {
  float a = acc + xb + emit;                 // FSUAdd, offset=0
  a = fminf(fmaxf(a, -256.0f), 256.0f);      // clamp to +/- 2^(depth-2)
  float o = (a >= 1.0f) ? 1.0f : 0.0f;       // emit bit when acc >= scale
  acc = a - o;
  emit = sr0 * o;                            // scrambled(sr[0]) AND out
  sr0 = 1.0f - o;                            // shift reg: sr[0] <- ~out
  return o;
}

__global__ __launch_bounds__(BLOCK) void fsusqrt_kernel(
    const float* __restrict__ x,          // [T, N] bits as f32
    const float* __restrict__ emit_init,  // [N]
    const float* __restrict__ acc_init,   // [N]
    const float* __restrict__ sr_init,    // [2, N] (only row 0 is live)
    float* __restrict__ out,              // [T, N]
    int N4, int T) {
  __shared__ v4f stage[DEPTH][BLOCK];     // 32 KB of 320 KB WGP LDS

  const int n4 = blockIdx.x * BLOCK + threadIdx.x;
  if (n4 >= N4) return;

  const v4f* x4 = (const v4f*)x;
  v4f* out4 = (v4f*)out;

  // Unpack carry state into scalar registers (sr[1] is dead state).
  v4f av = ((const v4f*)acc_init)[n4];
  v4f ev = ((const v4f*)emit_init)[n4];
  v4f sv = ((const v4f*)sr_init)[n4];  // row 0 of sr_init
  float a0 = av.x, a1 = av.y, a2 = av.z, a3 = av.w;
  float e0 = ev.x, e1 = ev.y, e2 = ev.z, e3 = ev.w;
  float s0 = sv.x, s1 = sv.y, s2 = sv.z, s3 = sv.w;

  // ---- prologue: fill DEPTH pipeline slots with async copies ----
  const int dlim = (T < DEPTH) ? T : DEPTH;
  for (int d = 0; d < dlim; ++d)
    async_copy_b128(lds_off(&stage[d][threadIdx.x]),
                    &x4[(size_t)d * N4 + n4]);

  int t = 0;
  // ---- steady state: wait oldest slot, consume, refill early, compute ----
  for (; t < T - DEPTH; ++t) {
    // async loads retire in order: <=DEPTH-1 outstanding => slot t is ready
    asm volatile("s_wait_asynccnt " STR(DEPTH_M1) ::: "memory");
    v4f xv = stage[t & (DEPTH - 1)][threadIdx.x];

    // WAR fence: async LDS writes are unordered vs DS reads from this wave.
    // Once the ds_load data is in registers, refill the slot immediately so
    // the next async load is in flight while we compute and store.
    asm volatile("s_wait_dscnt 0" ::: "memory");
    async_copy_b128(lds_off(&stage[t & (DEPTH - 1)][threadIdx.x]),
                    &x4[(size_t)(t + DEPTH) * N4 + n4]);

    v4f o;
    o.x = fsu_step(xv.x, a0, e0, s0);
    o.y = fsu_step(xv.y, a1, e1, s1);
    o.z = fsu_step(xv.z, a2, e2, s2);
    o.w = fsu_step(xv.w, a3, e3, s3);
    __builtin_nontemporal_store(o, &out4[(size_t)t * N4 + n4]);
  }

  // ---- drain: all remaining slots are in flight; wait for everything ----
  asm volatile("s_wait_asynccnt 0" ::: "memory");
  for (; t < T; ++t) {
    v4f xv = stage[t & (DEPTH - 1)][threadIdx.x];
    v4f o;
    o.x = fsu_step(xv.x, a0, e0, s0);
    o.y = fsu_step(xv.y, a1, e1, s1);
    o.z = fsu_step(xv.z, a2, e2, s2);
    o.w = fsu_step(xv.w, a3, e3, s3);
    __builtin_nontemporal_store(o, &out4[(size_t)t * N4 + n4]);
  }
}

extern "C" void kernel_launch(void* const* d_in, const int* in_sizes, int n_in,
                              void* d_out, int out_size, void* d_ws,
                              size_t ws_size, hipStream_t stream) {
  const float* x = (const float*)d_in[0];          // input   [T, N]
  const float* emit0 = (const float*)d_in[1];      // emit_init [N]
  const float* acc0 = (const float*)d_in[2];       // acc_init  [N]
  const float* sr0 = (const float*)d_in[3];        // sr_init [2, N]
  float* out = (float*)d_out;                      // outs    [T, N]

  const int N = in_sizes[1];
  const int T = in_sizes[0] / N;
  const int N4 = N / 4;
  const int grid = (N4 + BLOCK - 1) / BLOCK;

  fsusqrt_kernel<<<grid, BLOCK, 0, stream>>>(x, emit0, acc0, sr0, out, N4, T);
}